// DeepAttnBlock_3075196584117
// MI455X (gfx1250) — compile-verified
//
#include <hip/hip_runtime.h>
#include <math.h>

#define D 128
#define NEG_SLOPE 0.2f
#define LN_EPS 1e-5f

typedef float v2f __attribute__((ext_vector_type(2)));
typedef float v8f __attribute__((ext_vector_type(8)));

__device__ __forceinline__ float wave_sum(float v) {
#pragma unroll
  for (int off = 16; off > 0; off >>= 1) v += __shfl_xor(v, off, 32);
  return v;
}

// ---------------------------------------------------------------------------
// h = X @ W  via V_WMMA_F32_16X16X4_F32.  One block = 16 output rows,
// 8 waves -> 8 column tiles of 16.  A slab staged in LDS, W read from L2.
// ---------------------------------------------------------------------------
__global__ __launch_bounds__(256)
void gat_gemm_wmma(const float* __restrict__ x, const float* __restrict__ w,
                   float* __restrict__ h, int n_nodes) {
  __shared__ float lds_a[16 * D];  // 8 KB
  const int tid  = threadIdx.x;
  const int row0 = blockIdx.x * 16;

  for (int i = tid; i < 16 * D; i += 256) {
    int r = i >> 7, c = i & (D - 1);
    int row = row0 + r;
    lds_a[i] = (row < n_nodes) ? x[(size_t)row * D + c] : 0.0f;
  }
  __syncthreads();

  const int wave = tid >> 5;
  const int lane = tid & 31;
  const int half = lane >> 4;   // 0: K+{0,1}  1: K+{2,3}
  const int l16  = lane & 15;
  const int col0 = wave * 16;

  v8f c = {};
#pragma unroll
  for (int k = 0; k < D; k += 4) {
    const int ka = k + half * 2;
    v2f a, b;
    // A 16x4: lane = M (mod 16), VGPR0/1 = consecutive K within half
    a.x = lds_a[l16 * D + ka];
    a.y = lds_a[l16 * D + ka + 1];
    // B 4x16: lane = N (mod 16), rows K striped to match A halves
    b.x = w[ka * D + col0 + l16];
    b.y = w[(ka + 1) * D + col0 + l16];
    c = __builtin_amdgcn_wmma_f32_16x16x4_f32(false, a, false, b, (short)0, c,
                                              false, false);
  }
  // C/D 16x16 f32: VGPR v -> M = v (lanes 0-15) / v+8 (lanes 16-31), N = lane%16
#pragma unroll
  for (int v = 0; v < 8; ++v) {
    int row = row0 + v + half * 8;
    if (row < n_nodes) h[(size_t)row * D + col0 + l16] = c[v];
  }
}

// ---------------------------------------------------------------------------
// Per node: a_src = h.att_src, a_dst = h.att_dst, self-loop logit, and
// seed segment-max m[n] with the self-loop logit (every node has one).
// One wave per node, 4 floats per lane.
// ---------------------------------------------------------------------------
__global__ __launch_bounds__(256)
void gat_attn_scores(const float* __restrict__ h, const float* __restrict__ att_src,
                     const float* __restrict__ att_dst, float* __restrict__ a_src,
                     float* __restrict__ a_dst, float* __restrict__ self_logit,
                     float* __restrict__ mx, int n_nodes) {
  const int node = blockIdx.x * 8 + (threadIdx.x >> 5);
  const int lane = threadIdx.x & 31;
  if (node >= n_nodes) return;
  const float4 hv = ((const float4*)(h + (size_t)node * D))[lane];
  const float4 as = ((const float4*)att_src)[lane];
  const float4 ad = ((const float4*)att_dst)[lane];
  float ps = hv.x * as.x + hv.y * as.y + hv.z * as.z + hv.w * as.w;
  float pd = hv.x * ad.x + hv.y * ad.y + hv.z * ad.z + hv.w * ad.w;
  ps = wave_sum(ps);
  pd = wave_sum(pd);
  if (lane == 0) {
    a_src[node] = ps;
    a_dst[node] = pd;
    float lg = ps + pd;
    lg = lg > 0.0f ? lg : NEG_SLOPE * lg;
    self_logit[node] = lg;
    mx[node] = lg;
  }
}

// ---------------------------------------------------------------------------
// Per edge: logit = leakyrelu(a_src[s] + a_dst[d]); atomic max into m[d].
// ---------------------------------------------------------------------------
__global__ __launch_bounds__(256)
void gat_edge_logits(const int* __restrict__ ei, const float* __restrict__ a_src,
                     const float* __restrict__ a_dst, float* __restrict__ logit,
                     float* __restrict__ mx, int n_edges) {
  const int e = blockIdx.x * 256 + threadIdx.x;
  if (e >= n_edges) return;
  const int s = ei[e];
  const int d = ei[n_edges + e];
  float lg = a_src[s] + a_dst[d];
  lg = lg > 0.0f ? lg : NEG_SLOPE * lg;
  logit[e] = lg;
  __hip_atomic_fetch_max(&mx[d], lg, __ATOMIC_RELAXED, __HIP_MEMORY_SCOPE_AGENT);
}

// ---------------------------------------------------------------------------
// Per node: initialize acc with self-loop contribution and s with its weight.
// One thread per (node, 4-float chunk).
// ---------------------------------------------------------------------------
__global__ __launch_bounds__(256)
void gat_self_init(const float* __restrict__ h, const float* __restrict__ self_logit,
                   const float* __restrict__ mx, float* __restrict__ acc,
                   float* __restrict__ ssum, int n_nodes) {
  const int idx  = blockIdx.x * 256 + threadIdx.x;
  const int node = idx >> 5;
  const int c    = idx & 31;
  if (node >= n_nodes) return;
  const float e0 = __expf(self_logit[node] - mx[node]);
  float4 hv = ((const float4*)(h + (size_t)node * D))[c];
  float4 av = make_float4(hv.x * e0, hv.y * e0, hv.z * e0, hv.w * e0);
  ((float4*)(acc + (size_t)node * D))[c] = av;
  if (c == 0) ssum[node] = e0;
}

// ---------------------------------------------------------------------------
// Per edge (one wave): w = exp(logit - m[dst]); acc[dst] += w * h[src];
// ssum[dst] += w.  f32 atomics resolve in L2 (all node arrays are L2-resident).
// ---------------------------------------------------------------------------
__global__ __launch_bounds__(256)
void gat_edge_accum(const int* __restrict__ ei, const float* __restrict__ logit,
                    const float* __restrict__ mx, const float* __restrict__ h,
                    float* __restrict__ acc, float* __restrict__ ssum, int n_edges) {
  const int e    = blockIdx.x * 8 + (threadIdx.x >> 5);
  const int lane = threadIdx.x & 31;
  if (e >= n_edges) return;
  const int s = ei[e];
  const int d = ei[n_edges + e];
  const float w = __expf(logit[e] - mx[d]);
  const float4 hv = ((const float4*)(h + (size_t)s * D))[lane];
  float* ap = acc + (size_t)d * D + lane * 4;
  unsafeAtomicAdd(ap + 0, hv.x * w);
  unsafeAtomicAdd(ap + 1, hv.y * w);
  unsafeAtomicAdd(ap + 2, hv.z * w);
  unsafeAtomicAdd(ap + 3, hv.w * w);
  if (lane == 0) unsafeAtomicAdd(&ssum[d], w);
}

// ---------------------------------------------------------------------------
// Per node (one wave): y = x + acc/s + bias; LayerNorm; write out.
// ---------------------------------------------------------------------------
__global__ __launch_bounds__(256)
void gat_ln_out(const float* __restrict__ x, const float* __restrict__ acc,
                const float* __restrict__ ssum, const float* __restrict__ bias,
                const float* __restrict__ gamma, const float* __restrict__ beta,
                float* __restrict__ out, int n_nodes) {
  const int node = blockIdx.x * 8 + (threadIdx.x >> 5);
  const int lane = threadIdx.x & 31;
  if (node >= n_nodes) return;
  const float inv_s = 1.0f / ssum[node];
  const float4 xv = ((const float4*)(x   + (size_t)node * D))[lane];
  const float4 av = ((const float4*)(acc + (size_t)node * D))[lane];
  const float4 bv = ((const float4*)bias)[lane];
  float4 y;
  y.x = xv.x + av.x * inv_s + bv.x;
  y.y = xv.y + av.y * inv_s + bv.y;
  y.z = xv.z + av.z * inv_s + bv.z;
  y.w = xv.w + av.w * inv_s + bv.w;
  float sum = y.x + y.y + y.z + y.w;
  float sq  = y.x * y.x + y.y * y.y + y.z * y.z + y.w * y.w;
  sum = wave_sum(sum);
  sq  = wave_sum(sq);
  const float mu   = sum * (1.0f / D);
  const float var  = sq * (1.0f / D) - mu * mu;
  const float rstd = rsqrtf(var + LN_EPS);
  const float4 gv = ((const float4*)gamma)[lane];
  const float4 tv = ((const float4*)beta)[lane];
  float4 o;
  o.x = (y.x - mu) * rstd * gv.x + tv.x;
  o.y = (y.y - mu) * rstd * gv.y + tv.y;
  o.z = (y.z - mu) * rstd * gv.z + tv.z;
  o.w = (y.w - mu) * rstd * gv.w + tv.w;
  ((float4*)(out + (size_t)node * D))[lane] = o;
}

// ---------------------------------------------------------------------------
extern "C" void kernel_launch(void* const* d_in, const int* in_sizes, int n_in,
                              void* d_out, int out_size, void* d_ws, size_t ws_size,
                              hipStream_t stream) {
  const float* x0   = (const float*)d_in[0];
  const int*   ei   = (const int*)d_in[1];
  const float* W    = (const float*)d_in[2];
  const float* asrc = (const float*)d_in[3];
  const float* adst = (const float*)d_in[4];
  const float* bias = (const float*)d_in[5];
  const float* gam  = (const float*)d_in[6];
  const float* bet  = (const float*)d_in[7];
  float* out = (float*)d_out;

  const int n  = in_sizes[0] / D;
  const int e  = in_sizes[1] / 2;
  const int nl = in_sizes[2] / (D * D);

  float* ws    = (float*)d_ws;
  float* h     = ws; ws += (size_t)n * D;
  float* acc   = ws; ws += (size_t)n * D;
  float* va    = ws; ws += n;
  float* vb    = ws; ws += n;
  float* slf   = ws; ws += n;
  float* mx    = ws; ws += n;
  float* ssum  = ws; ws += n;
  float* lgt   = ws; ws += e;

  for (int l = 0; l < nl; ++l) {
    const float* xin = (l == 0) ? x0 : out;
    const float* Wl  = W + (size_t)l * D * D;
    gat_gemm_wmma<<<(n + 15) / 16, 256, 0, stream>>>(xin, Wl, h, n);
    gat_attn_scores<<<(n + 7) / 8, 256, 0, stream>>>(h, asrc + l * D, adst + l * D,
                                                     va, vb, slf, mx, n);
    gat_edge_logits<<<(e + 255) / 256, 256, 0, stream>>>(ei, va, vb, lgt, mx, e);
    gat_self_init<<<(n * 32 + 255) / 256, 256, 0, stream>>>(h, slf, mx, acc, ssum, n);
    gat_edge_accum<<<(e + 7) / 8, 256, 0, stream>>>(ei, lgt, mx, h, acc, ssum, e);
    gat_ln_out<<<(n + 7) / 8, 256, 0, stream>>>(xin, acc, ssum, bias + l * D,
                                                gam + l * D, bet + l * D, out, n);
  }
}